// binnings_CRVD_25864293056739
// MI455X (gfx1250) — compile-verified
//
#include <hip/hip_runtime.h>

typedef __attribute__((ext_vector_type(2))) float v2f;
typedef __attribute__((ext_vector_type(8))) float v8f;

#define HW   (512 * 512)
#define NCH  16
#define NG   4
#define ITERS 32                       // 32 pixels per wave-iteration
#define POS_PER_WAVE (32 * ITERS)      // 1024
#define WAVES_PER_B  (HW / POS_PER_WAVE) // 256
#define TOTAL_WAVES  (16 * WAVES_PER_B)  // 4096
#define BLOCK_THREADS 256
#define GRID_BLOCKS  (TOTAL_WAVES / (BLOCK_THREADS / 32)) // 512

// out[b,g,p] = sum_c Wm[g,c] * x[b,c,p], Wm[g,c] = ((c&3)==g) ? w[g][c>>2] : 0
// Mapped to chained V_WMMA_F32_16X16X4_F32 (exact fp32), K-chain over 4 channel
// groups, two 16-pixel N-tiles per iteration, NT loads/stores (stream-once,
// 256 MiB read > 192 MB L2).
__global__ __launch_bounds__(BLOCK_THREADS)
void binnings_wmma_kernel(const float* __restrict__ x,
                          const float* __restrict__ wgt,
                          float* __restrict__ out) {
  const int lane = threadIdx.x & 31;
  const int li   = lane & 15;
  const int half = lane >> 4;
  const int waveId = threadIdx.x >> 5;

  const int gw  = blockIdx.x * (BLOCK_THREADS / 32) + waveId;
  const int b   = gw / WAVES_PER_B;
  const int seg = gw % WAVES_PER_B;
  const long pbase = (long)seg * POS_PER_WAVE;

  // A matrices for the 4-step K-chain. 16x4 f32 A layout:
  // vgpr0: lanes 0-15 = (M=lane, K=0), lanes 16-31 = (M=lane-16, K=2)
  // vgpr1: lanes 0-15 = (M=lane, K=1), lanes 16-31 = (M=lane-16, K=3)
  // Nonzero A[m,kk] only at kk==m (m<4): lanes 0,1,18,19.
  v2f A[4];
#pragma unroll
  for (int k = 0; k < 4; ++k) {
    float a0 = 0.0f, a1 = 0.0f;
    if (lane == 0)  a0 = wgt[0 * 4 + k];   // (M=0,K=0)
    if (lane == 18) a0 = wgt[2 * 4 + k];   // (M=2,K=2)
    if (lane == 1)  a1 = wgt[1 * 4 + k];   // (M=1,K=1)
    if (lane == 19) a1 = wgt[3 * 4 + k];   // (M=3,K=3)
    A[k].x = a0;
    A[k].y = a1;
  }

  const float* xb = x + (long)b * NCH * HW;
  float* ob = out + (long)b * NG * HW;

  for (int it = 0; it < ITERS; ++it) {
    const long p0 = pbase + (long)it * 32;
    v8f acc0 = {};
    v8f acc1 = {};
#pragma unroll
    for (int k = 0; k < 4; ++k) {
      // B (4x16) layout: vgpr0: lanes0-15=(K=0,N=li), lanes16-31=(K=2,N=li)
      //                  vgpr1: lanes0-15=(K=1,N=li), lanes16-31=(K=3,N=li)
      const int ch0 = 4 * k + 2 * half;           // K row for vgpr0 in this lane half
      const float* c0 = xb + (long)ch0 * HW + p0 + li;
      const float* c1 = c0 + HW;                  // K row for vgpr1
      v2f b0, b1;
      b0.x = __builtin_nontemporal_load(c0);        // tile0 pixels p0..p0+15
      b0.y = __builtin_nontemporal_load(c1);
      b1.x = __builtin_nontemporal_load(c0 + 16);   // tile1 pixels p0+16..p0+31
      b1.y = __builtin_nontemporal_load(c1 + 16);
      acc0 = __builtin_amdgcn_wmma_f32_16x16x4_f32(false, A[k], false, b0,
                                                   (short)0, acc0, false, false);
      acc1 = __builtin_amdgcn_wmma_f32_16x16x4_f32(false, A[k], false, b1,
                                                   (short)0, acc1, false, false);
    }
    // D rows g=0..3 live in lanes 0-15 of acc vgprs 0..3 (lanes 16-31 hold
    // zero rows 8..11). Pull tile1's lanes 0-15 into lanes 16-31 with a
    // bpermute so each g-store is one full-wave 128B contiguous NT store.
#pragma unroll
    for (int g = 0; g < 4; ++g) {
      int hi = __builtin_amdgcn_ds_bpermute(li << 2,
                                            __builtin_bit_cast(int, acc1[g]));
      float v = half ? __builtin_bit_cast(float, hi) : acc0[g];
      __builtin_nontemporal_store(v, ob + (long)g * HW + p0 + lane);
    }
  }
}

extern "C" void kernel_launch(void* const* d_in, const int* in_sizes, int n_in,
                              void* d_out, int out_size, void* d_ws, size_t ws_size,
                              hipStream_t stream) {
  const float* x   = (const float*)d_in[0];   // (16, 16, 512, 512) fp32
  const float* wgt = (const float*)d_in[1];   // (4, 4, 1, 1) fp32
  float* out = (float*)d_out;                 // (16, 4, 512, 512) fp32
  (void)in_sizes; (void)n_in; (void)out_size; (void)d_ws; (void)ws_size;
  binnings_wmma_kernel<<<GRID_BLOCKS, BLOCK_THREADS, 0, stream>>>(x, wgt, out);
}